// MPNN_47115791237991
// MI455X (gfx1250) — compile-verified
//
#include <hip/hip_runtime.h>

typedef float v2f __attribute__((ext_vector_type(2)));
typedef float v8f __attribute__((ext_vector_type(8)));

static constexpr int B_DIM = 8;
static constexpr int NMAX  = 36;
static constexpr int HDIM  = 64;
static constexpr int STEPS = 5;
static constexpr int TILE_STRIDE = 68;            // 64 + 4 pad -> conflict-free b64 LDS reads
static constexpr int TILE_ELEMS  = HDIM * HDIM;   // 4096 floats per E tile

// Wave-relative LDS byte offset of a __shared__ pointer (low 32 bits of the
// generic address, see ISA 10.2).
__device__ __forceinline__ unsigned lds_off_of(const void* p) {
    return (unsigned)(unsigned long long)p;
}

// Async DMA one 16-byte chunk: global memory -> LDS, no VGPR payload.
// Tracked by ASYNCcnt (CDNA5 ISA).
__device__ __forceinline__ void async_g2lds_b128(unsigned lds_byte, const float* g) {
    asm volatile("global_load_async_to_lds_b128 %0, %1, off"
                 :: "v"(lds_byte), "v"(g)
                 : "memory");
}
__device__ __forceinline__ void wait_async0() {
    asm volatile("s_wait_asynccnt 0" ::: "memory");
}

// One workgroup per (b, j) output node. 256 threads = 8 wave32s.
// Waves: rt = wave&3 picks 16 output channels; lhalf = (wave>>2)*32 splits the
// l-reduction. Accumulators live in WMMA C registers across all 36 i-tiles.
__global__ __launch_bounds__(256) void mpnn_step_kernel(
    const float* __restrict__ edge,      // [B,N,N,H,H]
    const float* __restrict__ nodes_in,  // [B,N,H]
    float*       __restrict__ nodes_out, // [B,N,H]
    const float* __restrict__ mask,      // [B,N,1]
    const float* __restrict__ w_ih,      // [3H,H]
    const float* __restrict__ w_hh,      // [3H,H]
    const float* __restrict__ b_ih,      // [3H]
    const float* __restrict__ b_hh)      // [3H]
{
    __shared__ float tileA[2][HDIM * TILE_STRIDE]; // double-buffered E tile
    __shared__ float xs[NMAX * HDIM];              // all node states for this batch
    __shared__ float msgs[HDIM];
    __shared__ float gi_s[3 * HDIM];
    __shared__ float gh_s[3 * HDIM];

    const int tid  = threadIdx.x;
    const int lane = tid & 31;          // wave32
    const int wav  = tid >> 5;
    const int b    = blockIdx.x / NMAX;
    const int j    = blockIdx.x % NMAX;

    // stage all of nodes[b,:,:] into LDS (read repeatedly as WMMA B operands)
    for (int t = tid; t < NMAX * HDIM; t += 256)
        xs[t] = nodes_in[b * NMAX * HDIM + t];
    if (tid < HDIM) msgs[tid] = 0.0f;

    const float* gbase = edge + ((size_t)(b * NMAX) * NMAX + j) * TILE_ELEMS;
    const size_t tile_step = (size_t)NMAX * TILE_ELEMS; // i -> i+1 stride in floats

    // this thread's 4 x 16B slices of a tile (precomputed LDS byte offsets)
    const unsigned tbase0 = lds_off_of(&tileA[0][0]);
    const unsigned tbase1 = lds_off_of(&tileA[1][0]);
    unsigned lslice[4];
    int      gslice[4];
#pragma unroll
    for (int q = 0; q < 4; ++q) {
        const int c4  = tid + q * 256;            // float4 chunk id within tile
        const int row = c4 >> 4;                  // 16 chunks per 64-float row
        const int col = (c4 & 15) << 2;
        lslice[q] = (unsigned)((row * TILE_STRIDE + col) * 4);
        gslice[q] = c4 * 4;
    }
    auto issue_tile = [&](int i, int buf) {
        const float*   src   = gbase + (size_t)i * tile_step;
        const unsigned lbase = buf ? tbase1 : tbase0;
#pragma unroll
        for (int q = 0; q < 4; ++q)
            async_g2lds_b128(lbase + lslice[q], src + gslice[q]);
    };

    const int rt    = wav & 3;               // output-channel tile (16 rows)
    const int lhalf = (wav >> 2) << 5;       // 0 or 32: reduction split
    const int m     = lane & 15;             // A-matrix M index for this lane
    const int kk    = (lane >> 4) << 1;      // K sub-offset: lanes 0-15 -> K0/K1, 16-31 -> K2/K3

    v8f c = {};

    issue_tile(0, 0);                        // prologue DMA
    wait_async0();
    __syncthreads();                         // tile 0 + xs + msgs visible

    for (int i = 0; i < NMAX; ++i) {
        if (i + 1 < NMAX) issue_tile(i + 1, (i + 1) & 1);   // DMA overlaps WMMA below
        if (i + 2 < NMAX) {                                 // L2 prefetch of tile i+2
            const float* nxt2 = gbase + (size_t)(i + 2) * tile_step;
            __builtin_prefetch(nxt2 + (size_t)tid * 16, 0, 1);
        }

        const float* Ar = &tileA[i & 1][(rt * 16 + m) * TILE_STRIDE + lhalf + kk];
        const float* Xv = &xs[i * HDIM + lhalf + kk];
#pragma unroll
        for (int p = 0; p < 32; p += 4) {
            v2f a = *(const v2f*)(Ar + p);             // A[m, k] = E[kch, l]
            v2f x = *(const v2f*)(Xv + p);             // B[k, n] = x[l] broadcast to all n
            c = __builtin_amdgcn_wmma_f32_16x16x4_f32(
                    false, a, false, x, (short)0, c, false, false);
        }

        wait_async0();                                 // tile i+1 landed in LDS
        __syncthreads();                               // everyone done with tile i
    }

    // D[m,n] = y[m] in every column: lane 0 holds y[rt*16+v], lane 16 holds y[rt*16+8+v]
    if (lane == 0 || lane == 16) {
        const int off = rt * 16 + ((lane >> 4) << 3);
#pragma unroll
        for (int v = 0; v < 8; ++v)
            atomicAdd(&msgs[off + v], c[v]);           // ds_add_f32, 2 waves per slot
    }
    __syncthreads();
    if (tid < HDIM) msgs[tid] *= (1.0f / (float)(NMAX * NMAX));
    __syncthreads();

    // ---- fused GRU cell for node (b, j) ----
    if (tid < 3 * HDIM) {
        float ai = b_ih[tid];
        float ah = b_hh[tid];
        const float* wi = w_ih + tid * HDIM;
        const float* wh = w_hh + tid * HDIM;
        const float* h0 = &xs[j * HDIM];
#pragma unroll 8
        for (int l = 0; l < HDIM; ++l) {
            ai = fmaf(wi[l], msgs[l], ai);
            ah = fmaf(wh[l], h0[l], ah);
        }
        gi_s[tid] = ai;
        gh_s[tid] = ah;
    }
    __syncthreads();
    if (tid < HDIM) {
        const float r = 1.0f / (1.0f + expf(-(gi_s[tid] + gh_s[tid])));
        const float z = 1.0f / (1.0f + expf(-(gi_s[HDIM + tid] + gh_s[HDIM + tid])));
        const float n = tanhf(gi_s[2 * HDIM + tid] + r * gh_s[2 * HDIM + tid]);
        const float h = xs[j * HDIM + tid];
        const float out = ((1.0f - z) * n + z * h) * mask[b * NMAX + j];
        nodes_out[(size_t)(b * NMAX + j) * HDIM + tid] = out;
    }
}

extern "C" void kernel_launch(void* const* d_in, const int* in_sizes, int n_in,
                              void* d_out, int out_size, void* d_ws, size_t ws_size,
                              hipStream_t stream) {
    const float* edge  = (const float*)d_in[0];
    const float* nodes = (const float*)d_in[1];
    const float* mask  = (const float*)d_in[2];
    const float* w_ih  = (const float*)d_in[3];
    const float* w_hh  = (const float*)d_in[4];
    const float* b_ih  = (const float*)d_in[5];
    const float* b_hh  = (const float*)d_in[6];

    float* buf0 = (float*)d_ws;
    float* buf1 = buf0 + B_DIM * NMAX * HDIM;
    float* outp = (float*)d_out;

    const float* in_s = nodes;
    for (int s = 0; s < STEPS; ++s) {
        float* out_s = (s == STEPS - 1) ? outp : ((s & 1) ? buf1 : buf0);
        mpnn_step_kernel<<<B_DIM * NMAX, 256, 0, stream>>>(
            edge, in_s, out_s, mask, w_ih, w_hh, b_ih, b_hh);
        in_s = out_s;
    }
}